// FALayer_81862076662613
// MI455X (gfx1250) — compile-verified
//
#include <hip/hip_runtime.h>
#include <cstdint>
#include <cstddef>

// CDNA5 / gfx1250, wave32.
typedef float v2f __attribute__((ext_vector_type(2)));
typedef float v8f __attribute__((ext_vector_type(8)));

#define WAVE 32

// ---------------------------------------------------------------------------
// Kernel 1: zero the output accumulator (harness poisons d_out with 0xAA).
// ---------------------------------------------------------------------------
__global__ void fa_zero_kernel(float* __restrict__ z, int n) {
    int i = blockIdx.x * blockDim.x + threadIdx.x;
    if (i < n) z[i] = 0.0f;
}

// ---------------------------------------------------------------------------
// Kernel 2: gate scores via V_WMMA_F32_16X16X4_F32.
//   s_dst = emb @ gate_w[0:64], s_src = emb @ gate_w[64:128]
// One wave computes a 16-node tile: C(16x16) = A(16x64-chunked) x B(64x16),
// where B column 0 = w_dst, column 1 = w_src, columns 2..15 = 0.
// A fragment (ISA 7.12.2, 32-bit A 16x4): lanes 0-15 hold K pair {0,1},
// lanes 16-31 hold K pair {2,3}, M = lane%16. B mirrors with N = lane%16.
// C layout: VGPR v holds (M=v, N=lane) lanes 0-15, (M=v+8, N=lane-16) else.
// ---------------------------------------------------------------------------
__global__ void fa_scores_kernel(const float* __restrict__ emb,
                                 const float* __restrict__ gw,   // [128]
                                 float* __restrict__ sdst,       // [N]
                                 float* __restrict__ ssrc,       // [N]
                                 int n_nodes, int n_tiles) {
    int wave = (blockIdx.x * blockDim.x + threadIdx.x) / WAVE;
    int lane = threadIdx.x & (WAVE - 1);
    if (wave >= n_tiles) return;            // wave-uniform: EXEC stays all-1s

    const int row0 = wave * 16;
    const int col  = lane & 15;             // M for A-loads, N for B/C
    const int hi   = lane >> 4;             // 0: K pair {0,1}, 1: K pair {2,3}

    int row = row0 + col;
    if (row >= n_nodes) row = n_nodes - 1;  // clamp (keeps EXEC full); store guarded
    const float* __restrict__ erow = emb + (size_t)row * 64;

    const float mdst = (col == 0) ? 1.0f : 0.0f;
    const float msrc = (col == 1) ? 1.0f : 0.0f;

    v8f acc = {0.f, 0.f, 0.f, 0.f, 0.f, 0.f, 0.f, 0.f};

#pragma unroll
    for (int kb = 0; kb < 64; kb += 4) {
        const int k0 = kb + hi * 2;
        v2f a;
        a.x = erow[k0];
        a.y = erow[k0 + 1];
        v2f b;
        b.x = gw[k0]     * mdst + gw[64 + k0]     * msrc;
        b.y = gw[k0 + 1] * mdst + gw[64 + k0 + 1] * msrc;
        // (neg_a, A, neg_b, B, c_mod, C, reuse_a, reuse_b)
        acc = __builtin_amdgcn_wmma_f32_16x16x4_f32(false, a, false, b,
                                                    (short)0, acc, false, false);
    }

    // Extract: N==0 column -> s_dst, N==1 column -> s_src.
    float* __restrict__ outp = (col == 0) ? sdst : ((col == 1) ? ssrc : nullptr);
    if (outp) {
        const int base = row0 + hi * 8;
#pragma unroll
        for (int v = 0; v < 8; ++v) {
            const int r = base + v;
            if (r < n_nodes) outp[r] = acc[v];
        }
    }
}

// ---------------------------------------------------------------------------
// Kernel 3: edge gate + scatter-add, 32 edges per wave per iteration.
// Stage A (vectorized across lanes): lane i owns edge e0+i -> coalesced
//   src/dst loads, per-lane score/deg gathers, ONE tanh per edge.
// Stage B (unrolled 32x): readlane broadcasts (s,t,w) of edge i into SGPRs;
//   all 32 lanes gather emb[s] as one 256B burst (float2/lane) and issue
//   two FP32 global atomic adds into z[t] (L2-resident RMW).
// Tail: invalid lanes force w=0 (adding 0.0f is value-preserving), keeping
// the inner loop uniform and EXEC full.
// ---------------------------------------------------------------------------
__global__ void fa_scatter_kernel(const float* __restrict__ emb,
                                  const float* __restrict__ deg,
                                  const float* __restrict__ sdst,
                                  const float* __restrict__ ssrc,
                                  const float* __restrict__ gate_b, // [1]
                                  const int*   __restrict__ src,
                                  const int*   __restrict__ dst,
                                  float* __restrict__ z,
                                  int n_edges) {
    const int gid    = blockIdx.x * blockDim.x + threadIdx.x;
    const int wave   = gid / WAVE;
    const int lane   = threadIdx.x & (WAVE - 1);
    const int nwaves = (gridDim.x * blockDim.x) / WAVE;
    const int stride = nwaves * WAVE;           // edges per grid iteration
    const float bias = gate_b[0];

    for (int e0 = wave * WAVE; e0 < n_edges; e0 += stride) {
        // ---- Stage A: 32 edges' scalars, one per lane -------------------
        const int  e     = e0 + lane;
        const bool valid = (e < n_edges);
        const int  ei    = valid ? e : (n_edges - 1);

        // Prefetch next group's index streams (global_prefetch_b8).
        if (e0 + stride < n_edges) {
            __builtin_prefetch(src + e0 + stride + lane, 0, 1);
            __builtin_prefetch(dst + e0 + stride + lane, 0, 1);
        }

        const int s = src[ei];                  // coalesced
        const int t = dst[ei];                  // coalesced
        float w = tanhf(sdst[t] + ssrc[s] + bias) * deg[t] * deg[s];
        if (!valid) w = 0.0f;                   // tail-safe, branch-free

        // ---- Stage B: broadcast each edge, gather row, scatter-add ------
#pragma unroll
        for (int i = 0; i < WAVE; ++i) {
            const int   si = __builtin_amdgcn_readlane(s, i);   // SGPR
            const int   ti = __builtin_amdgcn_readlane(t, i);   // SGPR
            const float wi = __int_as_float(
                __builtin_amdgcn_readlane(__float_as_int(w), i));

            const v2f ev = *(const v2f*)(emb + (size_t)si * 64 + 2 * lane);
            float* __restrict__ zp = z + (size_t)ti * 64 + 2 * lane;
            unsafeAtomicAdd(zp,     ev.x * wi);  // global_atomic_add_f32
            unsafeAtomicAdd(zp + 1, ev.y * wi);
        }
    }
}

// ---------------------------------------------------------------------------
// Launcher. Input order: emb, deg, gate_w, gate_b, src, dst.
// d_out: z = N*64 floats. d_ws: s_dst[N] + s_src[N] floats.
// ---------------------------------------------------------------------------
extern "C" void kernel_launch(void* const* d_in, const int* in_sizes, int n_in,
                              void* d_out, int out_size, void* d_ws, size_t ws_size,
                              hipStream_t stream) {
    const float* emb    = (const float*)d_in[0];
    const float* deg    = (const float*)d_in[1];
    const float* gate_w = (const float*)d_in[2];
    const float* gate_b = (const float*)d_in[3];
    const int*   src    = (const int*)d_in[4];
    const int*   dst    = (const int*)d_in[5];
    float*       z      = (float*)d_out;

    const int N = in_sizes[1];          // deg has one entry per node
    const int E = in_sizes[4];

    float* sdst = (float*)d_ws;
    float* ssrc = sdst + N;

    // 1) zero z
    {
        const int threads = 256;
        const int blocks  = (out_size + threads - 1) / threads;
        fa_zero_kernel<<<blocks, threads, 0, stream>>>(z, out_size);
    }

    // 2) WMMA gate scores (one wave per 16-node tile, 8 waves per block)
    {
        const int n_tiles = (N + 15) / 16;
        const int threads = 256;                       // 8 waves
        const int blocks  = (n_tiles + 7) / 8;
        fa_scores_kernel<<<blocks, threads, 0, stream>>>(emb, gate_w, sdst, ssrc,
                                                         N, n_tiles);
    }

    // 3) edge scatter: 32 edges per wave-iteration, grid-stride
    {
        const int threads = 256;                       // 8 waves
        const int blocks  = 2048;                      // 16384 waves in flight
        fa_scatter_kernel<<<blocks, threads, 0, stream>>>(emb, deg, sdst, ssrc,
                                                          gate_b, src, dst, z, E);
    }
}